// NodeModel_90950227460159
// MI455X (gfx1250) — compile-verified
//
#include <hip/hip_runtime.h>

#define N_NODES 16384
#define N_EDGES 524288

typedef __attribute__((ext_vector_type(2))) float v2f;
typedef __attribute__((ext_vector_type(8))) float v8f;

// ---------------------------------------------------------------------------
// Step 1: z = (1 + eps) * x      (eps is a device scalar)
// ---------------------------------------------------------------------------
__global__ void init_z_kernel(const float* __restrict__ x,
                              const float* __restrict__ eps,
                              float* __restrict__ z) {
    int i = blockIdx.x * blockDim.x + threadIdx.x;
    if (i < N_NODES * 3) z[i] = (1.0f + eps[0]) * x[i];
}

// ---------------------------------------------------------------------------
// Step 2: z[tgt] += x[src] over all edges (scatter-add, f32 global atomics)
// edge_idx is int64 in the reference: row 0 = src, row 1 = tgt
// ---------------------------------------------------------------------------
__global__ void scatter_kernel(const float* __restrict__ x,
                               const long long* __restrict__ eidx,
                               float* __restrict__ z) {
    int e = blockIdx.x * blockDim.x + threadIdx.x;
    if (e >= N_EDGES) return;
    int s = (int)eidx[e];
    int t = (int)eidx[N_EDGES + e];
    atomicAdd(&z[t * 3 + 0], x[s * 3 + 0]);
    atomicAdd(&z[t * 3 + 1], x[s * 3 + 1]);
    atomicAdd(&z[t * 3 + 2], x[s * 3 + 2]);
}

// ---------------------------------------------------------------------------
// Step 3: h = relu(relu(z @ W1 + b1) @ W2 + b2)   (one thread per node;
// weight loads are wave-uniform -> scalar loads)
// ---------------------------------------------------------------------------
__global__ void mlp_kernel(const float* __restrict__ z,
                           const float* __restrict__ W1, const float* __restrict__ b1,
                           const float* __restrict__ W2, const float* __restrict__ b2,
                           float* __restrict__ h) {
    int i = blockIdx.x * blockDim.x + threadIdx.x;
    if (i >= N_NODES) return;
    float z0 = z[i * 3 + 0], z1 = z[i * 3 + 1], z2 = z[i * 3 + 2];
    float hid[16];
#pragma unroll
    for (int j = 0; j < 16; ++j) {
        float v = z0 * W1[0 * 16 + j] + z1 * W1[1 * 16 + j] + z2 * W1[2 * 16 + j] + b1[j];
        hid[j] = v > 0.0f ? v : 0.0f;
    }
#pragma unroll
    for (int j = 0; j < 16; ++j) {
        float v = b2[j];
#pragma unroll
        for (int k = 0; k < 16; ++k) v += hid[k] * W2[k * 16 + j];
        h[i * 16 + j] = v > 0.0f ? v : 0.0f;
    }
}

// ---------------------------------------------------------------------------
// Step 4: out = h @ h.T   (16384 x 16384 x 16, f32 WMMA 16x16x4, K chained x4)
//
// Fragment layouts (CDNA5 ISA 7.12.2, wave32):
//   A (16x4 f32): lane l: half=l>>4, m=l&15; v0 = A[m][2*half], v1 = A[m][2*half+1]
//   B (4x16 f32): lane l: n=l&15;  v0 = B[2*half][n],  v1 = B[2*half+1][n]
//   C/D (16x16 f32, 8 VGPRs): VGPR r -> row (r + 8*half), col = l&15
//
// For C[i][j] = sum_k h[i][k]*h[j][k]:  A tile from rows i0.., B tile is
// h.T -> B[k][n] = h[j0+n][k], so both fragments load h[row][2*half + k0 ...].
// Each wave computes 4 consecutive j-tiles reusing its A fragment.
// Output (1 GB) is streamed with non-temporal stores (won't fit 192MB L2).
// ---------------------------------------------------------------------------
__global__ __launch_bounds__(256) void gemm_hht_kernel(const float* __restrict__ h,
                                                       float* __restrict__ out) {
    const int lane = threadIdx.x & 31;
    const int wave = threadIdx.x >> 5;
    const int half = lane >> 4;
    const int mn   = lane & 15;

    const int i0     = blockIdx.y * 16;
    const int j0base = (blockIdx.x * 8 + wave) * 64;   // 4 tiles of 16 cols

    // A fragments for the 4 K-chunks (K = 16 total, 4 per WMMA)
    const float* arow = h + (size_t)(i0 + mn) * 16 + 2 * half;
    v2f a[4];
#pragma unroll
    for (int kc = 0; kc < 4; ++kc)
        a[kc] = *(const v2f*)(arow + 4 * kc);

#pragma unroll
    for (int t = 0; t < 4; ++t) {
        const int j0 = j0base + t * 16;
        const float* brow = h + (size_t)(j0 + mn) * 16 + 2 * half;

        v8f c = {0.0f, 0.0f, 0.0f, 0.0f, 0.0f, 0.0f, 0.0f, 0.0f};
#pragma unroll
        for (int kc = 0; kc < 4; ++kc) {
            v2f b = *(const v2f*)(brow + 4 * kc);
            // 8 args: (neg_a, A, neg_b, B, c_mod, C, reuse_a, reuse_b)
            c = __builtin_amdgcn_wmma_f32_16x16x4_f32(
                    false, a[kc], false, b, (short)0, c, false, false);
        }

        float* orow = out + (size_t)(i0 + 8 * half) * N_NODES + j0 + mn;
#pragma unroll
        for (int r = 0; r < 8; ++r)
            __builtin_nontemporal_store(c[r], orow + (size_t)r * N_NODES);
    }
}

// ---------------------------------------------------------------------------
extern "C" void kernel_launch(void* const* d_in, const int* in_sizes, int n_in,
                              void* d_out, int out_size, void* d_ws, size_t ws_size,
                              hipStream_t stream) {
    const float*     x    = (const float*)d_in[0];       // [N, 3]
    const long long* eidx = (const long long*)d_in[1];   // [2, E] int64
    const float*     W1   = (const float*)d_in[2];       // [3, 16]
    const float*     b1   = (const float*)d_in[3];       // [16]
    const float*     W2   = (const float*)d_in[4];       // [16, 16]
    const float*     b2   = (const float*)d_in[5];       // [16]
    const float*     eps  = (const float*)d_in[6];       // scalar
    float*           out  = (float*)d_out;               // [N, N]

    float* z = (float*)d_ws;                  // [N, 3]   (192 KB)
    float* h = z + (size_t)N_NODES * 3;       // [N, 16]  (1 MB), 16B-aligned

    init_z_kernel<<<(N_NODES * 3 + 255) / 256, 256, 0, stream>>>(x, eps, z);
    scatter_kernel<<<N_EDGES / 256, 256, 0, stream>>>(x, eidx, z);
    mlp_kernel<<<N_NODES / 256, 256, 0, stream>>>(z, W1, b1, W2, b2, h);
    // 1024 j-tiles / (8 waves * 4 tiles) = 32 blocks in x; 1024 i-tiles in y
    gemm_hht_kernel<<<dim3(32, 1024), 256, 0, stream>>>(h, out);
}